// ResidualVectorQuantizer_89094801588631
// MI455X (gfx1250) — compile-verified
//
#include <hip/hip_runtime.h>
#include <cstddef>

typedef __attribute__((ext_vector_type(2))) float v2f;
typedef __attribute__((ext_vector_type(4))) float v4f;
typedef __attribute__((ext_vector_type(8))) float v8f;

#define NUM_CB   8
#define CB_SIZE  256
#define DIM      128
#define RPW      16              // rows per wave
#define WPB      8               // waves per block
#define RPB      (RPW * WPB)     // 128 rows per block
#define CT       128             // codes per LDS tile (2 tiles per codebook)

// -------------------------------------------------------------------------
// Main RVQ kernel: one wave handles 16 rows; residual kept in LDS; the
// distance cross-term is computed with exact-precision f32 WMMA.
// -------------------------------------------------------------------------
__global__ __launch_bounds__(256, 1) void rvq_main(
    const float* __restrict__ z,
    const float* __restrict__ cb,
    float* __restrict__ q_out,
    int*   __restrict__ idx_out,
    float* __restrict__ blk_loss,
    int total_rows)
{
  __shared__ __align__(16) float cb_tile[CT * DIM];      // 64 KB
  __shared__ __align__(16) float res[WPB][RPW * DIM];    // 64 KB
  __shared__ float e_sq[CT];
  __shared__ int   idx_s[WPB][RPW];
  __shared__ float wave_loss[WPB];

  const int tid  = threadIdx.x;
  const int wave = tid >> 5;
  const int lane = tid & 31;
  const int half = lane >> 4;      // 0: lanes 0-15, 1: lanes 16-31
  const int l16  = lane & 15;
  const long wave_row0 = (long)blockIdx.x * RPB + (long)wave * RPW;
  (void)total_rows;

  // residual <- z  (16 rows x 128 f32 per wave, float4 vectorized)
  {
    const v4f* zr = (const v4f*)(z + wave_row0 * DIM);
    v4f* rr = (v4f*)&res[wave][0];
#pragma unroll
    for (int i = 0; i < 16; ++i)
      rr[lane + 32 * i] = zr[lane + 32 * i];
  }

  float loss_acc = 0.0f;

  for (int k = 0; k < NUM_CB; ++k) {
    // ---- Preload A fragments for this stage (residual fixed per stage).
    // f32 16x4 A layout: lane -> row l16; VGPR pair holds K = kb*4 + 2*half, +1.
    v2f afrag[32];
    {
      const v2f* ar = (const v2f*)&res[wave][l16 * DIM + 2 * half];
#pragma unroll
      for (int kb = 0; kb < 32; ++kb)
        afrag[kb] = ar[kb * 2];           // stride 4 floats = 2 v2f
    }

    float best_s[8];
    int   best_i[8];
#pragma unroll
    for (int v = 0; v < 8; ++v) { best_s[v] = 3.4028235e38f; best_i[v] = 0; }

    for (int ct = 0; ct < 2; ++ct) {
      __syncthreads();
      // ---- Stage 128 codes (64 KB) of codebook k into LDS cooperatively.
      {
        const v4f* src = (const v4f*)(cb + ((size_t)k * CB_SIZE + ct * CT) * DIM);
        v4f* dst = (v4f*)cb_tile;
#pragma unroll
        for (int i = 0; i < 16; ++i)
          dst[tid + 256 * i] = src[tid + 256 * i];
      }
      __syncthreads();
      // ---- e_sq per staged code (x_sq is row-constant -> irrelevant to argmin)
      if (tid < CT) {
        const float* c = &cb_tile[tid * DIM];
        float s = 0.0f;
#pragma unroll
        for (int d = 0; d < DIM; ++d) s = fmaf(c[d], c[d], s);
        e_sq[tid] = s;
      }
      __syncthreads();

      // ---- 8 N-tiles of 16 codes; 32 chained f32 WMMAs each (K = 128).
#pragma unroll
      for (int nt = 0; nt < 8; ++nt) {
        const int c_loc = nt * 16 + l16;                    // my column's code
        const v2f* br = (const v2f*)&cb_tile[c_loc * DIM + 2 * half];
        v8f acc = {};
#pragma unroll
        for (int kb = 0; kb < 32; ++kb) {
          acc = __builtin_amdgcn_wmma_f32_16x16x4_f32(
              false, afrag[kb], false, br[kb * 2], (short)0, acc, false, false);
        }
        const float es = e_sq[c_loc];
        const int cg = ct * CT + c_loc;
#pragma unroll
        for (int v = 0; v < 8; ++v) {
          // score = ||e||^2 - 2 r.e ; strict '<' keeps first (lowest) index
          float s = fmaf(-2.0f, acc[v], es);
          if (s < best_s[v]) { best_s[v] = s; best_i[v] = cg; }
        }
      }
    }

    // ---- Cross-lane argmin within each 16-lane half (row M = v + 8*half),
    //      tie-break to lower index to match jnp.argmin.
#pragma unroll
    for (int v = 0; v < 8; ++v) {
      float s = best_s[v];
      int   i = best_i[v];
#pragma unroll
      for (int m = 1; m < 16; m <<= 1) {
        float s2 = __shfl_xor(s, m, 32);
        int   i2 = __shfl_xor(i, m, 32);
        if (s2 < s || (s2 == s && i2 < i)) { s = s2; i = i2; }
      }
      best_i[v] = i;
    }
    // lanes 0..7 (half 0) publish rows 0..7; lanes 16..23 publish rows 8..15
    if (l16 < 8)
      idx_s[wave][l16 + 8 * half] = best_i[l16];
    __builtin_amdgcn_wave_barrier();   // intra-wave LDS ordering fence

    // ---- Emit indices for this stage
    if (half == 0)
      idx_out[(wave_row0 + l16) * NUM_CB + k] = idx_s[wave][l16];

    // ---- residual -= cb[k][idx];  loss += ||new residual||^2
    const float* cbk = cb + (size_t)k * CB_SIZE * DIM;
#pragma unroll
    for (int r = 0; r < RPW; ++r) {
      const int ci = idx_s[wave][r];
      v4f c4 = ((const v4f*)(cbk + (size_t)ci * DIM))[lane];
      v4f* rp = (v4f*)&res[wave][r * DIM];
      v4f nr = rp[lane] - c4;
      rp[lane] = nr;
      loss_acc += nr.x * nr.x + nr.y * nr.y + nr.z * nr.z + nr.w * nr.w;
    }
  }

  // ---- quantized = z + sg(quant_sum - z) == quant_sum == z - residual_final
  {
    const v4f* zr = (const v4f*)(z + wave_row0 * DIM);
    const v4f* rr = (const v4f*)&res[wave][0];
    v4f* qo = (v4f*)(q_out + wave_row0 * DIM);
#pragma unroll
    for (int i = 0; i < 16; ++i)
      qo[lane + 32 * i] = zr[lane + 32 * i] - rr[lane + 32 * i];
  }

  // ---- Deterministic loss reduction: wave shuffles -> LDS -> thread 0
#pragma unroll
  for (int m = 1; m < 32; m <<= 1)
    loss_acc += __shfl_xor(loss_acc, m, 32);
  if (lane == 0) wave_loss[wave] = loss_acc;
  __syncthreads();
  if (tid == 0) {
    float s = 0.0f;
#pragma unroll
    for (int w = 0; w < WPB; ++w) s += wave_loss[w];
    blk_loss[blockIdx.x] = s;
  }
}

// Final deterministic (fixed-order) sum of per-block partials.
__global__ void rvq_loss_final(const float* __restrict__ blk_loss, int nblk,
                               float* __restrict__ loss_out, float scale)
{
  if (threadIdx.x == 0 && blockIdx.x == 0) {
    float s = 0.0f;
    for (int i = 0; i < nblk; ++i) s += blk_loss[i];
    *loss_out = s * scale;
  }
}

extern "C" void kernel_launch(void* const* d_in, const int* in_sizes, int n_in,
                              void* d_out, int out_size, void* d_ws, size_t ws_size,
                              hipStream_t stream) {
  (void)n_in; (void)out_size; (void)ws_size;
  const float* z  = (const float*)d_in[0];   // [B,T,D] f32
  const float* cb = (const float*)d_in[1];   // [K,C,D] f32

  const int total_rows = in_sizes[0] / DIM;  // B*T = 262144
  const int nblk = total_rows / RPB;         // 2048

  float* q_out    = (float*)d_out;                                    // [B,T,D]
  int*   idx_out  = (int*)((float*)d_out + (size_t)total_rows * DIM); // [B,T,K] int32
  float* loss_out = (float*)d_out + (size_t)total_rows * DIM
                                  + (size_t)total_rows * NUM_CB;      // scalar
  float* blk_loss = (float*)d_ws;                                     // nblk floats

  rvq_main<<<nblk, 256, 0, stream>>>(z, cb, q_out, idx_out, blk_loss, total_rows);

  const float scale = 1.25f / ((float)total_rows * (float)DIM);
  rvq_loss_final<<<1, 32, 0, stream>>>(blk_loss, nblk, loss_out, scale);
}